// GraphAttention_9740985827789
// MI455X (gfx1250) — compile-verified
//
#include <hip/hip_runtime.h>

typedef float v2f __attribute__((ext_vector_type(2)));
typedef float v8f __attribute__((ext_vector_type(8)));

#define GB   4
#define GN   4096
#define GIN  256
#define GOUT 128
#define NT   (GN / 16)      // 256 row-tiles per batch
#define NWAVE 8
#define JCHUNK (GN / NWAVE) // 512 j-columns per wave
#define INV_TAU 0.5f
#define SLOPE   0.2f
#define MASK_NEG (-1e30f)

__device__ __forceinline__ v8f wmma_f32_16x16x4(v2f a, v2f b, v8f c) {
  // D = A(16x4) * B(4x16) + C(16x16), all f32, wave32
  return __builtin_amdgcn_wmma_f32_16x16x4_f32(false, a, false, b, (short)0, c,
                                               false, false);
}

// ---------------------------------------------------------------------------
// Kernel 1: h[b,i,:] = x[b,i,:] @ W   via f32 WMMA.
// Block = 8 waves; wave w computes the 16-wide column tile n0 = w*16 of a
// 16-row i-tile.
// ---------------------------------------------------------------------------
__global__ __launch_bounds__(256) void gat_h_kernel(const float* __restrict__ x,
                                                    const float* __restrict__ W,
                                                    float* __restrict__ h) {
  const int gid  = blockIdx.x;          // [0, GB*NT)
  const int b    = gid / NT;
  const int i0   = (gid % NT) * 16;
  const int wave = threadIdx.x >> 5;
  const int lane = threadIdx.x & 31;
  const int lh   = lane >> 4;           // lane half
  const int lm   = lane & 15;
  const int n0   = wave * 16;

  v8f acc;
#pragma unroll
  for (int v = 0; v < 8; ++v) acc[v] = 0.0f;

  const float* xrow = x + ((size_t)b * GN + i0 + lm) * GIN + lh * 2;
  const float* wcol = W + (size_t)(lh * 2) * GOUT + n0 + lm;

#pragma unroll 4
  for (int k = 0; k < GIN; k += 4) {
    // A-frag: (lane,v) -> x[i0+lm][k + lh*2 + v]
    const float2 xa = *reinterpret_cast<const float2*>(xrow + k);
    v2f a; a[0] = xa.x; a[1] = xa.y;
    // B-frag: (lane,v) -> W[k + lh*2 + v][n0 + lm]
    const float* wp = wcol + (size_t)k * GOUT;
    v2f bb; bb[0] = wp[0]; bb[1] = wp[GOUT];
    acc = wmma_f32_16x16x4(a, bb, acc);
  }

  // C/D layout: (lane,v) -> row i0 + lh*8 + v, col n0 + lm
  float* hout = h + ((size_t)b * GN + i0 + lh * 8) * GOUT + n0 + lm;
#pragma unroll
  for (int v = 0; v < 8; ++v) hout[(size_t)v * GOUT] = acc[v];
}

// ---------------------------------------------------------------------------
// Kernel 2: s[b,n] = h[b,n,:].a_src ; d[b,n] = h[b,n,:].a_dst
// ---------------------------------------------------------------------------
__global__ __launch_bounds__(256) void gat_sd_kernel(const float* __restrict__ h,
                                                     const float* __restrict__ a_src,
                                                     const float* __restrict__ a_dst,
                                                     float* __restrict__ s_ws,
                                                     float* __restrict__ d_ws) {
  const int idx   = blockIdx.x * 256 + threadIdx.x;   // < GB*GN
  const float* hr = h + (size_t)idx * GOUT;
  float s = 0.f, d = 0.f;
#pragma unroll 8
  for (int c = 0; c < GOUT; ++c) {
    const float v = hr[c];
    s = fmaf(v, a_src[c], s);
    d = fmaf(v, a_dst[c], d);
  }
  s_ws[idx] = s;
  d_ws[idx] = d;
}

// ---------------------------------------------------------------------------
// Kernel 3: fused masked-softmax attention.
// Block = one 16-row i-tile; each of the 8 waves owns a disjoint 512-wide
// j-chunk (scores computed exactly once; A streamed from HBM exactly once)
// and keeps the full 16x128 partial accumulator (8 WMMA tiles) plus online
// softmax stats (m,l) replicated in lane pairs (r, r+16). Partials are merged
// across waves through LDS at the end (flash-attention combine).
// ---------------------------------------------------------------------------
__global__ __launch_bounds__(256) void gat_attn_kernel(const float* __restrict__ Am,
                                                       const float* __restrict__ h,
                                                       const float* __restrict__ s_ws,
                                                       const float* __restrict__ d_ws,
                                                       float* __restrict__ out) {
  __shared__ float d_sh[GN];                    // 16 KB: d[b,:]
  __shared__ float lds_acc[NWAVE][16 * GOUT];   // 64 KB: partial acc tiles
  __shared__ float lds_m[NWAVE][16];
  __shared__ float lds_l[NWAVE][16];

  const int gid  = blockIdx.x;
  const int b    = gid / NT;
  const int i0   = (gid % NT) * 16;
  const int wave = threadIdx.x >> 5;
  const int lane = threadIdx.x & 31;
  const int lh   = lane >> 4;
  const int lm   = lane & 15;

  for (int j = threadIdx.x; j < GN; j += 256) d_sh[j] = d_ws[b * GN + j];
  __syncthreads();

  const float s_row = s_ws[b * GN + i0 + lm];   // score source term, row lm
  const float* Arow = Am + ((size_t)b * GN + i0 + lm) * GN + lh * 2;
  const int jbeg = wave * JCHUNK;
  const int jend = jbeg + JCHUNK;

  v8f acc[8];
#pragma unroll
  for (int t = 0; t < 8; ++t)
#pragma unroll
    for (int v = 0; v < 8; ++v) acc[t][v] = 0.0f;
  float m = MASK_NEG;
  float l = 0.f;

  for (int j0 = jbeg; j0 < jend; j0 += 16) {
    __builtin_prefetch(Arow + j0 + 16, 0, 1);   // next A tile of this row

    float e[4][2];
    float p[4][2];
    float tmax = MASK_NEG;
#pragma unroll
    for (int kk = 0; kk < 4; ++kk) {
      // lane's mask values: A[i0+lm][j0 + kk*4 + lh*2 + {0,1}]
      const float2 av = *reinterpret_cast<const float2*>(Arow + j0 + kk * 4);
#pragma unroll
      for (int v = 0; v < 2; ++v) {
        const float aval = (v == 0) ? av.x : av.y;
        const float dv   = d_sh[j0 + kk * 4 + lh * 2 + v];
        float t = s_row + dv;
        t = (t > 0.f) ? t : t * SLOPE;          // leaky_relu
        t *= INV_TAU;                            // score / tau
        t = (aval >= 1e-9f) ? t : MASK_NEG;      // mask
        e[kk][v] = t;
        tmax = fmaxf(tmax, t);
      }
    }
    // full 16-column row max: other half of the row lives in lane^16
    tmax = fmaxf(tmax, __shfl_xor(tmax, 16, 32));
    const float m_new = fmaxf(m, tmax);
    const float scale = __expf(m - m_new);       // ==1 when max unchanged

    float rowsum = 0.f;
#pragma unroll
    for (int kk = 0; kk < 4; ++kk)
#pragma unroll
      for (int v = 0; v < 2; ++v) {
        const float pe = (e[kk][v] <= -1e20f) ? 0.f : __expf(e[kk][v] - m_new);
        p[kk][v] = pe;
        rowsum += pe;
      }
    rowsum += __shfl_xor(rowsum, 16, 32);
    l = l * scale + rowsum;                      // exact when scale==1

    // rescale accumulators only when some row max actually moved
    if (__any(m_new != m)) {
#pragma unroll
      for (int v = 0; v < 8; ++v) {
        const float sc = __shfl(scale, lh * 8 + v, 32);  // scale of acc row
#pragma unroll
        for (int t = 0; t < 8; ++t) acc[t][v] *= sc;
      }
    }
    m = m_new;

    // acc += P(16x16) @ H(16x128); A-fragment reused across the 8 n-tiles
#pragma unroll
    for (int kk = 0; kk < 4; ++kk) {
      v2f pa; pa[0] = p[kk][0]; pa[1] = p[kk][1];
      const float* hb = h + ((size_t)b * GN + j0 + kk * 4 + lh * 2) * GOUT + lm;
#pragma unroll
      for (int t = 0; t < 8; ++t) {
        v2f hbf; hbf[0] = hb[16 * t]; hbf[1] = hb[16 * t + GOUT];
        acc[t] = wmma_f32_16x16x4(pa, hbf, acc[t]);
      }
    }
  }

  // ---- publish partials -------------------------------------------------
  if (lane < 16) {                // lanes 0..15 hold (m,l) for rows 0..15
    lds_m[wave][lm] = m;
    lds_l[wave][lm] = l;
  }
#pragma unroll
  for (int t = 0; t < 8; ++t)
#pragma unroll
    for (int v = 0; v < 8; ++v)
      lds_acc[wave][(lh * 8 + v) * GOUT + 16 * t + lm] = acc[t][v];
  __syncthreads();

  // ---- cross-wave softmax merge: thread -> (row r, 8-col group) ----------
  const int r  = threadIdx.x >> 4;          // 0..15
  const int c0 = (threadIdx.x & 15) * 8;    // 0..120
  float M = MASK_NEG;
#pragma unroll
  for (int w = 0; w < NWAVE; ++w) M = fmaxf(M, lds_m[w][r]);
  float wgt[NWAVE];
  float L = 0.f;
#pragma unroll
  for (int w = 0; w < NWAVE; ++w) {
    const float ew = __expf(lds_m[w][r] - M);   // 0 for all-masked chunks
    wgt[w] = ew;
    L += lds_l[w][r] * ew;
  }
  const float invL = (M > -1e20f) ? (1.f / L) : 0.f;  // nan_to_num path

  float* op = out + ((size_t)b * GN + i0 + r) * GOUT + c0;
#pragma unroll
  for (int c = 0; c < 8; ++c) {
    float sum = 0.f;
#pragma unroll
    for (int w = 0; w < NWAVE; ++w)
      sum = fmaf(lds_acc[w][r * GOUT + c0 + c], wgt[w], sum);
    op[c] = sum * invL;
  }
}

// ---------------------------------------------------------------------------
extern "C" void kernel_launch(void* const* d_in, const int* in_sizes, int n_in,
                              void* d_out, int out_size, void* d_ws, size_t ws_size,
                              hipStream_t stream) {
  (void)in_sizes; (void)n_in; (void)out_size; (void)ws_size;
  const float* x     = (const float*)d_in[0];   // [B,N,256]
  const float* Amask = (const float*)d_in[1];   // [B,N,N]
  const float* W     = (const float*)d_in[2];   // [256,128]
  const float* a_src = (const float*)d_in[3];   // [128]
  const float* a_dst = (const float*)d_in[4];   // [128]
  float* out = (float*)d_out;                   // [B,N,128]

  float* h    = (float*)d_ws;                          // 8 MB
  float* s_ws = h + (size_t)GB * GN * GOUT;            // 64 KB
  float* d_vs = s_ws + (size_t)GB * GN;                // 64 KB

  gat_h_kernel<<<GB * NT, 256, 0, stream>>>(x, W, h);
  gat_sd_kernel<<<(GB * GN) / 256, 256, 0, stream>>>(h, a_src, a_dst, s_ws, d_vs);
  gat_attn_kernel<<<GB * NT, 256, 0, stream>>>(Amask, h, s_ws, d_vs, out);
}